// OriginalSPDMatrix_14963666059944
// MI455X (gfx1250) — compile-verified
//
#include <hip/hip_runtime.h>

// ---------------------------------------------------------------------------
// Fused SPD-matrix head for MI455X (gfx1250, wave32).
//   h        = softplus(x @ W1^T + b1)          [B,128]   (kept in LDS)
//   elements = h @ W2^T + b2                     [B,2080]  (kept in LDS)
//   L        = tril_scatter(elements)            (never materialized)
//   A        = L @ L^T + eps*I                   [B,64,64] (written to HBM)
// One workgroup (8 wave32s) owns 16 batch rows; all matmuls use
// V_WMMA_F32_16X16X4_F32 (exact f32). Intermediates never touch HBM.
// Softplus uses the branch-free stable form max(z,0)+log(1+exp(-|z|)) on the
// hardware transcendental units (v_exp_f32 / v_log_f32) instead of the ~40-op
// libm log1pf expansion.
// ---------------------------------------------------------------------------

typedef __attribute__((ext_vector_type(2))) float v2f;
typedef __attribute__((ext_vector_type(8))) float v8f;

#define B_TOT   16384
#define IN      128
#define HID     128
#define NUM_EL  2080
#define NTILES2 130          // 2080 / 16
#define NMAT    64
#define EPS_VAL 1e-6f

__device__ __forceinline__ float softplus_f(float z) {
    // Stable, branch-free: softplus(z) = max(z,0) + log1p(exp(-|z|)).
    // t = exp(-|z|) in (0,1]; 1+t in (1,2] where hardware log is accurate.
    const float t = __expf(-__builtin_fabsf(z));   // v_exp_f32
    return __builtin_fmaxf(z, 0.0f) + __logf(1.0f + t);  // v_log_f32
}

// D = A(16x4) * B(4x16) + C using the f32 WMMA path.
__device__ __forceinline__ v8f wmma_f32_k4(v2f a, v2f b, v8f c) {
    return __builtin_amdgcn_wmma_f32_16x16x4_f32(
        /*neg_a=*/false, a, /*neg_b=*/false, b,
        /*c_mod=*/(short)0, c, /*reuse_a=*/false, /*reuse_b=*/false);
}

extern "C" __global__ __launch_bounds__(256)
void spd_fused_kernel(const float* __restrict__ x,
                      const float* __restrict__ W1,
                      const float* __restrict__ b1,
                      const float* __restrict__ W2,
                      const float* __restrict__ b2,
                      float* __restrict__ out)
{
    extern __shared__ float lds[];
    float* h_lds = lds;                  // [16][HID]    =   8 KB
    float* els   = lds + 16 * HID;       // [16][NUM_EL] = 133 KB  (needs CDNA5 320KB LDS)

    const int tid  = threadIdx.x;
    const int lane = tid & 31;
    const int wave = tid >> 5;           // 8 wave32s per block
    const int l16  = lane & 15;
    const int lh   = lane >> 4;          // 0: lanes 0-15, 1: lanes 16-31
    const int b0   = blockIdx.x * 16;    // 16 batch rows per workgroup

    // ------------------------------------------------------------------
    // Phase 1: h = softplus(x @ W1^T + b1). Wave w computes cols [16w,16w+16).
    // A frag (16x4 f32): row = l16, k = k0 + 2*lh, pair in the 2 VGPRs.
    // B frag (4x16):     B[k][n] = W1[n][k] -> same contiguous-pair access.
    // ------------------------------------------------------------------
    {
        const int n0 = wave * 16;
        const float* xrow = x  + (size_t)(b0 + l16) * IN;
        const float* wrow = W1 + (size_t)(n0 + l16) * IN;
        v8f c = {};
        #pragma unroll 8
        for (int k0 = 0; k0 < IN; k0 += 4) {
            const int k = k0 + 2 * lh;
            v2f a;  a.x  = xrow[k];  a.y  = xrow[k + 1];
            v2f bb; bb.x = wrow[k];  bb.y = wrow[k + 1];
            c = wmma_f32_k4(a, bb, c);
        }
        const float bias = b1[n0 + l16];
        #pragma unroll
        for (int r = 0; r < 8; ++r) {
            const int m = r + 8 * lh;                 // C/D layout: M = r + 8*hi
            h_lds[m * HID + n0 + l16] = softplus_f(c[r] + bias);
        }
    }
    __syncthreads();

    // ------------------------------------------------------------------
    // Phase 2: elements = h @ W2^T + b2 -> LDS. 130 col-tiles over 8 waves.
    // W2 (1 MB) is L2-resident across all 1024 workgroups.
    // ------------------------------------------------------------------
    for (int t = wave; t < NTILES2; t += 8) {
        const int n0 = t * 16;
        if (t + 8 < NTILES2) {
            // warm L2/WGP$ for the next tile's W2 rows (global_prefetch_b8)
            __builtin_prefetch(W2 + (size_t)((t + 8) * 16 + l16) * HID, 0, 3);
        }
        const float* hrow = h_lds + l16 * HID;
        const float* wrow = W2 + (size_t)(n0 + l16) * HID;
        v8f c = {};
        #pragma unroll 8
        for (int k0 = 0; k0 < HID; k0 += 4) {
            const int k = k0 + 2 * lh;
            v2f a;  a.x  = hrow[k];  a.y  = hrow[k + 1];
            v2f bb; bb.x = wrow[k];  bb.y = wrow[k + 1];
            c = wmma_f32_k4(a, bb, c);
        }
        const float bias = b2[n0 + l16];
        #pragma unroll
        for (int r = 0; r < 8; ++r) {
            const int m = r + 8 * lh;
            els[m * NUM_EL + n0 + l16] = c[r] + bias;
        }
    }
    __syncthreads();

    // ------------------------------------------------------------------
    // Phase 3: A = L @ L^T + eps*I.  A[i][j] = sum_k L[i][k]*L[j][k].
    // L[r][k] = (k<=r) ? els[tri(r)+k] : 0  -- scatter folded into frag loads.
    // 16 batch x 16 (4x4) tiles = 256 wave-jobs; 32 per wave.
    // ------------------------------------------------------------------
    #pragma unroll 1
    for (int i = 0; i < 32; ++i) {
        const int job = wave + 8 * i;          // 0..255
        const int bi  = job >> 4;              // batch element in tile
        const int ti  = (job & 15) >> 2;       // A tile row
        const int tj  = job & 3;               // A tile col
        const float* e = els + bi * NUM_EL;

        const int ra   = ti * 16 + l16;        // L row feeding the A fragment
        const int rb   = tj * 16 + l16;        // L row feeding the B fragment
        const int triA = (ra * (ra + 1)) >> 1;
        const int triB = (rb * (rb + 1)) >> 1;

        v8f c = {};
        #pragma unroll 4
        for (int k0 = 0; k0 < NMAT; k0 += 4) {
            const int k = k0 + 2 * lh;
            v2f a, bb;
            a.x  = (k     <= ra) ? e[triA + k]     : 0.0f;
            a.y  = (k + 1 <= ra) ? e[triA + k + 1] : 0.0f;
            bb.x = (k     <= rb) ? e[triB + k]     : 0.0f;
            bb.y = (k + 1 <= rb) ? e[triB + k + 1] : 0.0f;
            c = wmma_f32_k4(a, bb, c);
        }

        float* orow   = out + (size_t)(b0 + bi) * (NMAT * NMAT);
        const int col = tj * 16 + l16;
        #pragma unroll
        for (int r = 0; r < 8; ++r) {
            const int row = ti * 16 + r + 8 * lh;
            orow[row * NMAT + col] = c[r] + ((row == col) ? EPS_VAL : 0.0f);
        }
    }
}

extern "C" void kernel_launch(void* const* d_in, const int* in_sizes, int n_in,
                              void* d_out, int out_size, void* d_ws, size_t ws_size,
                              hipStream_t stream) {
    (void)in_sizes; (void)n_in; (void)d_ws; (void)ws_size; (void)out_size;
    const float* x  = (const float*)d_in[0];
    const float* W1 = (const float*)d_in[1];
    const float* b1 = (const float*)d_in[2];
    const float* W2 = (const float*)d_in[3];
    const float* b2 = (const float*)d_in[4];
    float* out = (float*)d_out;

    // Dynamic LDS: 16*128 (h) + 16*2080 (elements) floats = 141,312 B.
    // Fits CDNA5's 320KB-per-workgroup LDS; one workgroup per WGP.
    const size_t shmem = (size_t)(16 * HID + 16 * NUM_EL) * sizeof(float);
    spd_fused_kernel<<<B_TOT / 16, 256, shmem, stream>>>(x, W1, b1, W2, b2, out);
}